// QuantumLayer_1116691497586
// MI455X (gfx1250) — compile-verified
//
#include <hip/hip_runtime.h>
#include <math.h>

typedef __attribute__((ext_vector_type(2))) float v2f;
typedef __attribute__((ext_vector_type(8))) float v8f;

#define NQ 4
#define NL 6
#define DIM 16

// ---------------------------------------------------------------------------
// Kernel 1: build Mmat[q][i][j] = A_q[i][j] = Re( U^H Z_q U )[i][j]
// where U is the fixed 16x16 unitary of the entangler layers.
// One block of 256 threads; threads 0..15 each evolve one identity column.
// ---------------------------------------------------------------------------
__global__ void build_Amat(const float* __restrict__ w, float* __restrict__ Mmat) {
  __shared__ float Ure[DIM][DIM];
  __shared__ float Uim[DIM][DIM];
  const int t = threadIdx.x;
  if (t < DIM) {
    const int j = t;
    float ar[DIM], ai[DIM];
#pragma unroll
    for (int k = 0; k < DIM; ++k) { ar[k] = (k == j) ? 1.f : 0.f; ai[k] = 0.f; }
#pragma unroll
    for (int l = 0; l < NL; ++l) {
      // RX(w[l][q]) on each wire. Basis index i = sum_q bit_q * 2^(3-q).
#pragma unroll
      for (int q = 0; q < NQ; ++q) {
        float s, c;
        __sincosf(0.5f * w[l * NQ + q], &s, &c);
        const int mask = 1 << (3 - q);
#pragma unroll
        for (int i = 0; i < DIM; ++i) {
          if (i & mask) continue;
          const int i1 = i | mask;
          const float a_r = ar[i], a_i = ai[i], b_r = ar[i1], b_i = ai[i1];
          // a' = c*a + (-i s)*b ; b' = (-i s)*a + c*b
          ar[i]  = c * a_r + s * b_i;
          ai[i]  = c * a_i - s * b_r;
          ar[i1] = c * b_r + s * a_i;
          ai[i1] = c * b_i - s * a_r;
        }
      }
      // CNOT ring: control q -> target (q+1)%4
#pragma unroll
      for (int q = 0; q < NQ; ++q) {
        const int cm = 1 << (3 - q);
        const int tm = 1 << (3 - ((q + 1) % NQ));
#pragma unroll
        for (int i = 0; i < DIM; ++i) {
          if ((i & cm) && !(i & tm)) {
            const int i1 = i | tm;
            float tr = ar[i]; ar[i] = ar[i1]; ar[i1] = tr;
            float ti = ai[i]; ai[i] = ai[i1]; ai[i1] = ti;
          }
        }
      }
    }
#pragma unroll
    for (int k = 0; k < DIM; ++k) { Ure[k][j] = ar[k]; Uim[k][j] = ai[k]; }
  }
  __syncthreads();
  // thread t = (i,j): A_q[i][j] = sum_k z_q(k) * Re(conj(U[k][i]) U[k][j])
  const int i = t >> 4, j = t & 15;
#pragma unroll
  for (int q = 0; q < NQ; ++q) {
    const int bit = 3 - q;
    float acc = 0.f;
#pragma unroll
    for (int k = 0; k < DIM; ++k) {
      const float zk = ((k >> bit) & 1) ? -1.f : 1.f;
      acc += zk * (Ure[k][i] * Ure[k][j] + Uim[k][i] * Uim[k][j]);
    }
    Mmat[q * 256 + i * 16 + j] = acc;
  }
}

// ---------------------------------------------------------------------------
// Kernel 2: per 16-sample tile, out[s][q] = v_s^T A_q v_s via fp32 WMMA.
//   Y_q = A_q * V^T  (A-frag = constant A_q chunk, B-frag = V^T chunk)
//   D layout: vgpr r, lane l -> Y_q[r + 8*(l/16)][l%16]
//   => lane l holds y-elements of its own sample (l%16); final dot is
//      8 FMAs + one shfl_xor(16).
// Lane-dependent v elements are built by substituting per-lane factors
// (f2 for the B-frag's qubit-2 bit, f0 for the dot's qubit-0 bit) so no
// register array / select trees are ever materialized.
// ---------------------------------------------------------------------------
__global__ __launch_bounds__(256) void qlayer_wmma(
    const float* __restrict__ x, const float* __restrict__ Mmat,
    float* __restrict__ out, int batch) {
  const int lane = threadIdx.x & 31;
  const bool hi  = (lane & 16) != 0; // lanes 16-31
  const int row  = lane & 15;        // sample-in-tile; also A-matrix row m
  const int half = hi ? 1 : 0;

  // Constant A-matrix fragments: element (m, k) of chunk c lives at
  // lane = m + 16*(k>>1), vgpr = k&1  ->  this lane needs A_q[row][4c+2*half+p]
  v2f afrag[4][4];
#pragma unroll
  for (int q = 0; q < 4; ++q)
#pragma unroll
    for (int c = 0; c < 4; ++c)
      afrag[q][c] = *(const v2f*)(Mmat + q * 256 + row * 16 + 4 * c + 2 * half);

  const int waveId = (int)((blockIdx.x * blockDim.x + threadIdx.x) >> 5);
  const int nwaves = (int)((gridDim.x * blockDim.x) >> 5);
  const int ntiles = batch >> 4;

  for (int tile = waveId; tile < ntiles; tile += nwaves) {
    const int sample = (tile << 4) + row;
    const float4 xv = *(const float4*)(x + (size_t)sample * 4);
    float s0, c0, s1, c1, s2, c2, s3, c3;
    __sincosf(0.5f * xv.x, &s0, &c0);
    __sincosf(0.5f * xv.y, &s1, &c1);
    __sincosf(0.5f * xv.z, &s2, &c2);
    __sincosf(0.5f * xv.w, &s3, &c3);

    // Per-lane substituted factors (1 cndmask each):
    const float f2 = hi ? s2 : c2;  // qubit-2 bit == half for B-frag elements
    const float f0 = hi ? s0 : c0;  // qubit-0 bit == half for dot elements

    // B-frag chunk c supplies v[k], k = 4c + 2*half + p:
    //   bits (b0,b1) = c, b2 = half, b3 = p
    const float a0 = c0 * c1, a1 = c0 * s1, a2 = s0 * c1, a3 = s0 * s1;
    const float h0 = a0 * f2, h1 = a1 * f2, h2 = a2 * f2, h3 = a3 * f2;
    v2f bfrag[4];
    bfrag[0].x = h0 * c3; bfrag[0].y = h0 * s3;
    bfrag[1].x = h1 * c3; bfrag[1].y = h1 * s3;
    bfrag[2].x = h2 * c3; bfrag[2].y = h2 * s3;
    bfrag[3].x = h3 * c3; bfrag[3].y = h3 * s3;

    // Dot elements vh[r] = v[8*half + r]: b0 = half, (b1,b2,b3) = r
    const float e0 = f0 * c1, e1 = f0 * s1;
    const float g0 = e0 * c2, g1 = e0 * s2, g2 = e1 * c2, g3 = e1 * s2;
    float vh[8];
    vh[0] = g0 * c3; vh[1] = g0 * s3; vh[2] = g1 * c3; vh[3] = g1 * s3;
    vh[4] = g2 * c3; vh[5] = g2 * s3; vh[6] = g3 * c3; vh[7] = g3 * s3;

    float res[4];
#pragma unroll
    for (int q = 0; q < 4; ++q) {
      v8f d = {0.f, 0.f, 0.f, 0.f, 0.f, 0.f, 0.f, 0.f};
#pragma unroll
      for (int c = 0; c < 4; ++c) {
        d = __builtin_amdgcn_wmma_f32_16x16x4_f32(
                false, afrag[q][c], false, bfrag[c], (short)0, d, false, false);
      }
      float partial = 0.f;
#pragma unroll
      for (int r = 0; r < 8; ++r) partial += d[r] * vh[r];
      res[q] = partial + __shfl_xor(partial, 16, 32);
    }
    if (!hi) {
      *(float4*)(out + (size_t)sample * 4) = make_float4(res[0], res[1], res[2], res[3]);
    }
  }
}

// ---------------------------------------------------------------------------
extern "C" void kernel_launch(void* const* d_in, const int* in_sizes, int n_in,
                              void* d_out, int out_size, void* d_ws, size_t ws_size,
                              hipStream_t stream) {
  const float* x = (const float*)d_in[0];       // [B,4] float32
  const float* w = (const float*)d_in[1];       // [6,4] float32
  float* out = (float*)d_out;                   // [B,4] float32
  float* Mmat = (float*)d_ws;                   // 4*16*16 floats scratch

  const int batch = in_sizes[0] / 4;

  build_Amat<<<1, 256, 0, stream>>>(w, Mmat);
  qlayer_wmma<<<1024, 256, 0, stream>>>(x, Mmat, out, batch);
}